// KDQuantizer_32126355375012
// MI455X (gfx1250) — compile-verified
//
#include <hip/hip_runtime.h>
#include <hip/hip_bf16.h>

typedef float v2f __attribute__((ext_vector_type(2)));
typedef float v8f __attribute__((ext_vector_type(8)));

#define B_SZ   16384
#define D_SZ   32
#define K_SZ   256
#define DIN    64
#define ROW_STRIDE (D_SZ * DIN)                    // 2048 floats between consecutive b for fixed d
#define CODES_N   ((size_t)B_SZ * D_SZ)            // 524288
#define OUTS_N    ((size_t)B_SZ * D_SZ * DIN)      // 33554432
#define REG_OFF   (CODES_N + OUTS_N)

__global__ void kdq_zero_reg(float* reg) { *reg = 0.0f; }

__global__ __launch_bounds__(256) void kdq_main(
    const float* __restrict__ x,    // (B, D, 64)
    const float* __restrict__ ck,   // (D, 256, 64)
    const float* __restrict__ cv,   // (D, 256, 64)
    float* __restrict__ out)        // [codes (B*D) | outputs (B*D*64) | reg (1)] as float
{
    const int lane      = threadIdx.x & 31;
    const int waveInBlk = threadIdx.x >> 5;
    const int w  = blockIdx.x * 8 + waveInBlk;     // 16384 waves total
    const int d  = w >> 9;                         // 512 row-blocks per d (consecutive waves share d)
    const int b0 = (w & 511) << 5;                 // 32 rows per wave

    const int nsub  = lane & 15;                   // row (A) / column (B) within tile
    const int khalf = lane >> 4;                   // K split across lane halves

    const float* xbase = x  + ((size_t)b0 * D_SZ + d) * DIN;
    const float* ckd   = ck + (size_t)d * K_SZ * DIN;
    const float* cvd   = cv + (size_t)d * K_SZ * DIN;

    // ---- A fragments (ISA 32-bit A 16x4 layout): lane<16 -> K {0,1}, lane>=16 -> K {2,3}
    // RT loads: each x row is re-read once in the gather phase (WGP$/L2 hit).
    v2f a0[16], a1[16];
#pragma unroll
    for (int kb = 0; kb < 16; ++kb) {
        const int ko = kb * 4 + khalf * 2;
        a0[kb] = *(const v2f*)(xbase + (size_t)nsub        * ROW_STRIDE + ko);
        a1[kb] = *(const v2f*)(xbase + (size_t)(16 + nsub) * ROW_STRIDE + ko);
    }

    float best0[8], best1[8];
    int   idx0[8],  idx1[8];
#pragma unroll
    for (int r = 0; r < 8; ++r) {
        best0[r] = -3.4e38f; best1[r] = -3.4e38f; idx0[r] = 0; idx1[r] = 0;
    }

    // ---- 16 column tiles of 16 centroids each
    for (int t = 0; t < 16; ++t) {
        v8f acc0 = {0,0,0,0,0,0,0,0};
        v8f acc1 = {0,0,0,0,0,0,0,0};
        float nrm = 0.f;
        const float* bcol = ckd + (size_t)(t * 16 + nsub) * DIN + khalf * 2;
#pragma unroll
        for (int kb = 0; kb < 16; ++kb) {
            v2f bf = *(const v2f*)(bcol + kb * 4);   // B 4x16 frag: lane<16 K{0,1}, lane>=16 K{2,3}
            nrm = fmaf(bf.x, bf.x, nrm);
            nrm = fmaf(bf.y, bf.y, nrm);
            acc0 = __builtin_amdgcn_wmma_f32_16x16x4_f32(false, a0[kb], false, bf,
                                                         (short)0, acc0, false, false);
            acc1 = __builtin_amdgcn_wmma_f32_16x16x4_f32(false, a1[kb], false, bf,
                                                         (short)0, acc1, false, false);
        }
        // complete ||c||^2 across the two lane-halves, fold into bias
        nrm += __shfl_xor(nrm, 16, 32);
        const float bias = -0.5f * nrm;
        const int   col  = t * 16 + nsub;
#pragma unroll
        for (int r = 0; r < 8; ++r) {
            float v = acc0[r] + bias;
            if (v > best0[r]) { best0[r] = v; idx0[r] = col; }
            v = acc1[r] + bias;
            if (v > best1[r]) { best1[r] = v; idx1[r] = col; }
        }
    }

    // ---- argmax butterfly within each 16-lane half (ties -> lower index)
#pragma unroll
    for (int r = 0; r < 8; ++r) {
#pragma unroll
        for (int off = 1; off < 16; off <<= 1) {
            float ov = __shfl_xor(best0[r], off, 32);
            int   oi = __shfl_xor(idx0[r],  off, 32);
            if (ov > best0[r] || (ov == best0[r] && oi < idx0[r])) { best0[r] = ov; idx0[r] = oi; }
            ov = __shfl_xor(best1[r], off, 32);
            oi = __shfl_xor(idx1[r],  off, 32);
            if (ov > best1[r] || (ov == best1[r] && oi < idx1[r])) { best1[r] = ov; idx1[r] = oi; }
        }
    }

    // ---- gather centroids_v rows, write outputs + codes, accumulate reg
    float* outsOut = out + CODES_N;
    float regacc = 0.f;
#pragma unroll
    for (int m = 0; m < 32; ++m) {
        const int r   = m & 7;
        const int src = (m & 8) ? 16 : 0;          // row m lives in VGPR r of half (m>>3)&1
        const int code = (m < 16) ? __shfl(idx0[r], src, 32)
                                  : __shfl(idx1[r], src, 32);
        const float* vrow = cvd   + (size_t)code * DIN;            // L2-resident, RT
        const float* xrow = xbase + (size_t)m * ROW_STRIDE;        // last use of this row -> NT
        v2f vv = *(const v2f*)(vrow + lane * 2);
        v2f xv = __builtin_nontemporal_load((const v2f*)(xrow + lane * 2));
        const float dx = vv.x - xv.x;
        const float dy = vv.y - xv.y;
        regacc = fmaf(dx, dx, regacc);
        regacc = fmaf(dy, dy, regacc);
        // write-once 134 MB stream -> NT store, don't pollute L2
        __builtin_nontemporal_store(vv,
            (v2f*)(outsOut + ((size_t)(b0 + m) * D_SZ + d) * DIN + lane * 2));
        if (lane == 0)
            __builtin_nontemporal_store((float)code,
                out + (size_t)(b0 + m) * D_SZ + d);
    }

    // ---- reg: wave reduce + single f32 atomic
#pragma unroll
    for (int off = 1; off < 32; off <<= 1) regacc += __shfl_xor(regacc, off, 32);
    if (lane == 0)
        atomicAdd(out + REG_OFF, regacc * (1.0f / ((float)B_SZ * (float)D_SZ * (float)DIN)));
}

extern "C" void kernel_launch(void* const* d_in, const int* in_sizes, int n_in,
                              void* d_out, int out_size, void* d_ws, size_t ws_size,
                              hipStream_t stream) {
    const float* x  = (const float*)d_in[0];   // inputs      (B, D, 64)
    const float* ck = (const float*)d_in[1];   // centroids_k (D, 256, 64)
    const float* cv = (const float*)d_in[2];   // centroids_v (D, 256, 64)
    float* out = (float*)d_out;

    kdq_zero_reg<<<1, 1, 0, stream>>>(out + REG_OFF);
    kdq_main<<<2048, 256, 0, stream>>>(x, ck, cv, out);
}